// HeteroGNN_90623809946067
// MI455X (gfx1250) — compile-verified
//
#include <hip/hip_runtime.h>

typedef __attribute__((ext_vector_type(2))) float v2f;
typedef __attribute__((ext_vector_type(8))) float v8f;

#define N_LIG  100000
#define N_TGT  20000
#define NEDGE  250000
#define LIG_IN 4
#define TGT_IN 1280
#define H      128

// ---- workspace layout (units: floats). Zero-initialized region comes first. ----
#define OFF_AGGL   0u          // [N_LIG, H]      12,800,000
#define OFF_AGG4   12800000u   // [N_TGT, 4]          80,000
#define OFF_CNTT   12880000u   // [N_TGT]             20,000
#define OFF_CNTL   12900000u   // [N_LIG]            100,000
#define ZERO_TOTAL 13000000u
#define OFF_YT     13000000u   // [N_TGT, H]       2,560,000   x_target @ W_tl_l
#define OFF_TR     15560000u   // [N_TGT, H]       2,560,000   x_target @ W_lt_r
#define OFF_SL     18120000u   // [N_LIG]            100,000
#define OFF_ST     18220000u   // [N_TGT]             20,000
#define OFF_BP0    18240000u   // packed W_tl_l      163,840
#define OFF_BP1    18403840u   // packed W_lt_r      163,840

__device__ __forceinline__ void atomicAddF(float* p, float v) {
    __hip_atomic_fetch_add(p, v, __ATOMIC_RELAXED, __HIP_MEMORY_SCOPE_AGENT);
}

// ---------------- zero the atomic-accumulated regions ----------------
__global__ __launch_bounds__(256) void zero_kernel(float4* __restrict__ p, int n4) {
    int i = blockIdx.x * 256 + threadIdx.x;
    if (i < n4) p[i] = make_float4(0.f, 0.f, 0.f, 0.f);
}

// ---- pack W[1280,128] into WMMA B-fragment layout: Bp[k/4][half][n] = {W[k+2h][n], W[k+2h+1][n]} ----
__global__ __launch_bounds__(256) void pack_kernel(const float* __restrict__ W0,
                                                   const float* __restrict__ W1,
                                                   v2f* __restrict__ Bp0, v2f* __restrict__ Bp1) {
    int i = blockIdx.x * 256 + threadIdx.x;          // over (1280/4)*2*128 = 81920
    if (i >= (TGT_IN / 4) * 2 * H) return;
    int n = i & 127;
    int h = (i >> 7) & 1;
    int t = i >> 8;
    int k = t * 4 + h * 2;
    v2f a; a.x = W0[k * H + n]; a.y = W0[(k + 1) * H + n];
    v2f b; b.x = W1[k * H + n]; b.y = W1[(k + 1) * H + n];
    Bp0[i] = a;
    Bp1[i] = b;
}

// ---------------- edge pass A: 4-dim ligand scatter + degree counts ----------------
__global__ __launch_bounds__(256) void edge_count_kernel(const int* __restrict__ src,
                                                         const int* __restrict__ dst,
                                                         const float* __restrict__ xl,
                                                         float* __restrict__ agg4,
                                                         float* __restrict__ cnt_t,
                                                         float* __restrict__ cnt_l) {
    int e = blockIdx.x * 256 + threadIdx.x;
    if (e >= NEDGE) return;
    int s = src[e], d = dst[e];
    float4 x = *(const float4*)(xl + (size_t)s * 4);
    atomicAddF(&agg4[d * 4 + 0], x.x);
    atomicAddF(&agg4[d * 4 + 1], x.y);
    atomicAddF(&agg4[d * 4 + 2], x.z);
    atomicAddF(&agg4[d * 4 + 3], x.w);
    atomicAddF(&cnt_t[d], 1.0f);
    atomicAddF(&cnt_l[s], 1.0f);
}

// ---------------- fused dual GEMM: Y0 = X @ W_tl_l, Y1 = X @ W_lt_r (fp32 WMMA) ----------------
// block = 256 threads = 8 waves; block owns a 32-row M tile (two 16-row WMMA tiles);
// wave w owns N columns [16w,16w+16). Each wave runs FOUR independent accumulator chains
// (2 M tiles x 2 weight matrices) reusing each B fragment twice. The 32xKC A panel is
// staged into LDS with double-buffered GLOBAL_LOAD_ASYNC_TO_LDS_B128 (ASYNCcnt) so the
// next chunk's DMA overlaps this chunk's WMMAs.
__global__ __launch_bounds__(256) void gemm_dual_kernel(const float* __restrict__ X,
                                                        const v2f* __restrict__ Bp0,
                                                        const v2f* __restrict__ Bp1,
                                                        float* __restrict__ Y0,
                                                        float* __restrict__ Y1) {
    constexpr int KC   = 128;                 // K chunk
    constexpr int ROWS = 32;                  // M rows per block
    __shared__ float As[2][ROWS * KC];        // 2 x 16 KB
    const int m0   = blockIdx.x * ROWS;
    const int tid  = threadIdx.x;
    const int lane = tid & 31;
    const int wave = tid >> 5;
    const int mrow = lane & 15;
    const int hsel = lane >> 4;               // 0: K pair {0,1}, 1: K pair {2,3}
    const int ncol = wave * 16 + mrow;

    // issue async DMA of one 32xKC chunk into buffer `buf` (1024 float4s, 4 per thread)
    auto issue = [&](int kc, int buf) {
        #pragma unroll
        for (int j = 0; j < 4; ++j) {
            int i = tid + j * 256;
            int r = i >> 5;                   // KC/4 = 32 float4 per row
            int c = (i & 31) << 2;
            const float* g = &X[(size_t)(m0 + r) * TGT_IN + kc + c];
            unsigned lds = (unsigned)(size_t)&As[buf][r * KC + c];
            asm volatile("global_load_async_to_lds_b128 %0, %1, off"
                         :: "v"(lds), "v"(g) : "memory");
        }
    };

    issue(0, 0);
    v8f c00 = {}, c01 = {}, c10 = {}, c11 = {};
    for (int kc = 0; kc < TGT_IN; kc += KC) {
        const int buf = (kc / KC) & 1;
        asm volatile("s_wait_asynccnt 0x0" ::: "memory");
        __syncthreads();                      // chunk `buf` visible to all waves; all waves
                                              // are also done computing on buf^1
        if (kc + KC < TGT_IN) issue(kc + KC, buf ^ 1);
        const float* A0 = &As[buf][0];
        #pragma unroll 4
        for (int k0 = 0; k0 < KC; k0 += 4) {
            v2f a0 = *(const v2f*)&A0[mrow * KC + k0 + hsel * 2];
            v2f a1 = *(const v2f*)&A0[(16 + mrow) * KC + k0 + hsel * 2];
            int bidx = ((kc + k0) >> 2) * 256 + hsel * 128 + ncol;
            v2f b0 = Bp0[bidx];
            v2f b1 = Bp1[bidx];
            c00 = __builtin_amdgcn_wmma_f32_16x16x4_f32(false, a0, false, b0, (short)0, c00, false, false);
            c01 = __builtin_amdgcn_wmma_f32_16x16x4_f32(false, a0, false, b1, (short)0, c01, false, false);
            c10 = __builtin_amdgcn_wmma_f32_16x16x4_f32(false, a1, false, b0, (short)0, c10, false, false);
            c11 = __builtin_amdgcn_wmma_f32_16x16x4_f32(false, a1, false, b1, (short)0, c11, false, false);
        }
    }
    const int mbase = m0 + hsel * 8;          // C/D layout: VGPR r -> row r (+8 for hi lanes)
    #pragma unroll
    for (int r = 0; r < 8; ++r) {
        Y0[(size_t)(mbase + r) * H + ncol]      = c00[r];
        Y1[(size_t)(mbase + r) * H + ncol]      = c01[r];
        Y0[(size_t)(mbase + 16 + r) * H + ncol] = c10[r];
        Y1[(size_t)(mbase + 16 + r) * H + ncol] = c11[r];
    }
}

// ---------------- edge pass B: scatter projected 128-dim target msgs onto ligands ----------------
__global__ __launch_bounds__(256) void scatter_kernel(const int* __restrict__ src,
                                                      const int* __restrict__ dst,
                                                      const float* __restrict__ yt,
                                                      float* __restrict__ aggL) {
    int e = blockIdx.x * 2 + (threadIdx.x >> 7);
    int f = threadIdx.x & 127;
    if (e >= NEDGE) return;
    int s = src[e], d = dst[e];
    float v = yt[(size_t)d * H + f];
    atomicAddF(&aggL[(size_t)s * H + f], v);
}

// ---------------- target finalize: t_out = relu(agg4/cnt @ W_lt_l + b + t_r); sT = t_out . W_ep_hi ----
__global__ __launch_bounds__(256) void target_final_kernel(const float* __restrict__ agg4,
                                                           const float* __restrict__ cnt_t,
                                                           const float* __restrict__ tr,
                                                           const float* __restrict__ W_lt_l,
                                                           const float* __restrict__ b_lt_l,
                                                           const float* __restrict__ W_ep,
                                                           float* __restrict__ sT) {
    int lane = threadIdx.x & 31;
    int t = blockIdx.x * 8 + (threadIdx.x >> 5);
    if (t >= N_TGT) return;
    float inv = 1.0f / fmaxf(cnt_t[t], 1.0f);
    float4 a4 = *(const float4*)&agg4[t * 4];
    a4.x *= inv; a4.y *= inv; a4.z *= inv; a4.w *= inv;
    int f = lane * 4;
    float4 h  = *(const float4*)&tr[(size_t)t * H + f];
    float4 b  = *(const float4*)&b_lt_l[f];
    float4 w0 = *(const float4*)&W_lt_l[0 * H + f];
    float4 w1 = *(const float4*)&W_lt_l[1 * H + f];
    float4 w2 = *(const float4*)&W_lt_l[2 * H + f];
    float4 w3 = *(const float4*)&W_lt_l[3 * H + f];
    float4 v;
    v.x = fmaxf(h.x + b.x + a4.x * w0.x + a4.y * w1.x + a4.z * w2.x + a4.w * w3.x, 0.f);
    v.y = fmaxf(h.y + b.y + a4.x * w0.y + a4.y * w1.y + a4.z * w2.y + a4.w * w3.y, 0.f);
    v.z = fmaxf(h.z + b.z + a4.x * w0.z + a4.y * w1.z + a4.z * w2.z + a4.w * w3.z, 0.f);
    v.w = fmaxf(h.w + b.w + a4.x * w0.w + a4.y * w1.w + a4.z * w2.w + a4.w * w3.w, 0.f);
    float4 we = *(const float4*)&W_ep[H + f];
    float s = v.x * we.x + v.y * we.y + v.z * we.z + v.w * we.w;
    #pragma unroll
    for (int off = 16; off > 0; off >>= 1) s += __shfl_xor(s, off, 32);
    if (lane == 0) sT[t] = s;
}

// ---------------- ligand finalize: l_out = relu(aggL/cnt + b + x_l @ W_tl_r); sL = l_out . W_ep_lo ----
__global__ __launch_bounds__(256) void ligand_final_kernel(const float* __restrict__ aggL,
                                                           const float* __restrict__ cnt_l,
                                                           const float* __restrict__ xl,
                                                           const float* __restrict__ W_tl_r,
                                                           const float* __restrict__ b_tl_l,
                                                           const float* __restrict__ W_ep,
                                                           float* __restrict__ sL) {
    int lane = threadIdx.x & 31;
    int l = blockIdx.x * 8 + (threadIdx.x >> 5);
    if (l >= N_LIG) return;
    float inv = 1.0f / fmaxf(cnt_l[l], 1.0f);
    float4 x = *(const float4*)&xl[(size_t)l * 4];
    int f = lane * 4;
    float4 h  = *(const float4*)&aggL[(size_t)l * H + f];
    h.x *= inv; h.y *= inv; h.z *= inv; h.w *= inv;
    float4 b  = *(const float4*)&b_tl_l[f];
    float4 w0 = *(const float4*)&W_tl_r[0 * H + f];
    float4 w1 = *(const float4*)&W_tl_r[1 * H + f];
    float4 w2 = *(const float4*)&W_tl_r[2 * H + f];
    float4 w3 = *(const float4*)&W_tl_r[3 * H + f];
    float4 v;
    v.x = fmaxf(h.x + b.x + x.x * w0.x + x.y * w1.x + x.z * w2.x + x.w * w3.x, 0.f);
    v.y = fmaxf(h.y + b.y + x.x * w0.y + x.y * w1.y + x.z * w2.y + x.w * w3.y, 0.f);
    v.z = fmaxf(h.z + b.z + x.x * w0.z + x.y * w1.z + x.z * w2.z + x.w * w3.z, 0.f);
    v.w = fmaxf(h.w + b.w + x.x * w0.w + x.y * w1.w + x.z * w2.w + x.w * w3.w, 0.f);
    float4 we = *(const float4*)&W_ep[f];
    float s = v.x * we.x + v.y * we.y + v.z * we.z + v.w * we.w;
    #pragma unroll
    for (int off = 16; off > 0; off >>= 1) s += __shfl_xor(s, off, 32);
    if (lane == 0) sL[l] = s;
}

// ---------------- final edge scores ----------------
__global__ __launch_bounds__(256) void edge_out_kernel(const int* __restrict__ src,
                                                       const int* __restrict__ dst,
                                                       const float* __restrict__ sL,
                                                       const float* __restrict__ sT,
                                                       const float* __restrict__ b_ep,
                                                       float* __restrict__ out) {
    int e = blockIdx.x * 256 + threadIdx.x;
    if (e < NEDGE) out[e] = sL[src[e]] + sT[dst[e]] + b_ep[0];
}

extern "C" void kernel_launch(void* const* d_in, const int* in_sizes, int n_in,
                              void* d_out, int out_size, void* d_ws, size_t ws_size,
                              hipStream_t stream) {
    const float* x_ligand = (const float*)d_in[0];
    const float* x_target = (const float*)d_in[1];
    const int*   edge_src = (const int*)d_in[2];
    const int*   edge_dst = (const int*)d_in[3];
    const float* W_lt_l   = (const float*)d_in[4];
    const float* b_lt_l   = (const float*)d_in[5];
    const float* W_lt_r   = (const float*)d_in[6];
    const float* W_tl_l   = (const float*)d_in[7];
    const float* b_tl_l   = (const float*)d_in[8];
    const float* W_tl_r   = (const float*)d_in[9];
    const float* W_ep     = (const float*)d_in[10];
    const float* b_ep     = (const float*)d_in[11];

    float* ws   = (float*)d_ws;
    float* aggL = ws + OFF_AGGL;
    float* agg4 = ws + OFF_AGG4;
    float* cntT = ws + OFF_CNTT;
    float* cntL = ws + OFF_CNTL;
    float* yt   = ws + OFF_YT;
    float* tr   = ws + OFF_TR;
    float* sL   = ws + OFF_SL;
    float* sT   = ws + OFF_ST;
    v2f*   Bp0  = (v2f*)(ws + OFF_BP0);
    v2f*   Bp1  = (v2f*)(ws + OFF_BP1);

    zero_kernel<<<(ZERO_TOTAL / 4 + 255) / 256, 256, 0, stream>>>((float4*)ws, ZERO_TOTAL / 4);
    pack_kernel<<<(81920 + 255) / 256, 256, 0, stream>>>(W_tl_l, W_lt_r, Bp0, Bp1);
    edge_count_kernel<<<(NEDGE + 255) / 256, 256, 0, stream>>>(edge_src, edge_dst, x_ligand,
                                                               agg4, cntT, cntL);
    gemm_dual_kernel<<<N_TGT / 32, 256, 0, stream>>>(x_target, Bp0, Bp1, yt, tr);
    scatter_kernel<<<NEDGE / 2, 256, 0, stream>>>(edge_src, edge_dst, yt, aggL);
    target_final_kernel<<<N_TGT / 8, 256, 0, stream>>>(agg4, cntT, tr, W_lt_l, b_lt_l, W_ep, sT);
    ligand_final_kernel<<<N_LIG / 8, 256, 0, stream>>>(aggL, cntL, x_ligand, W_tl_r, b_tl_l, W_ep, sL);
    edge_out_kernel<<<(NEDGE + 255) / 256, 256, 0, stream>>>(edge_src, edge_dst, sL, sT, b_ep,
                                                             (float*)d_out);
}